// GAT_4587025072294
// MI455X (gfx1250) — compile-verified
//
#include <hip/hip_runtime.h>
#include <hip/hip_bf16.h>

typedef float v2f __attribute__((ext_vector_type(2)));
typedef float v8f __attribute__((ext_vector_type(8)));

#define NHID 128
#define NEG_ATT 0.2f
#define NEG_ACT 0.01f

#if __has_builtin(__builtin_amdgcn_wmma_f32_16x16x4_f32)
#define HAS_WMMA_F32X4 1
#endif

// ---------------------------------------------------------------------------
// GEMM: H[nrows x 128] = X[nrows x 128] @ W[128 x 128]
// One workgroup = 256 threads = 8 waves; each wave computes a 16x128 strip
// with V_WMMA_F32_16X16X4_F32 (fp32-exact matrix path on CDNA5).
// W staged in LDS (64 KB) in pair-interleaved layout so each B fragment is
// one ds_load_b64 into an aligned VGPR pair:
//   Ws[(k>>1)*256 + n*2 + (k&1)] = W[k][n]
// ---------------------------------------------------------------------------
__global__ __launch_bounds__(256) void gat_gemm128(
    const float* __restrict__ X, const float* __restrict__ W,
    float* __restrict__ H, int nrows) {
  __shared__ float Ws[NHID * NHID];   // 64 KB, fragment-major
  const int tid = threadIdx.x;
  for (int i = tid; i < NHID * NHID; i += 256) {
    const int k = i >> 7, n = i & 127;
    Ws[((k >> 1) << 8) + (n << 1) + (k & 1)] = W[i];
  }
  __syncthreads();

  const int wave = tid >> 5;
  const int lane = tid & 31;
  const int row0 = blockIdx.x * 128 + wave * 16;
  if (row0 >= nrows) return;          // after the only barrier: safe

  const int m     = lane & 15;        // M (A) / N (B,C,D) within tile
  const int khv   = lane >> 4;        // which K pair: 0 -> K0,1 ; 1 -> K2,3
  const int khalf = khv * 2;
  const int mhi   = khv * 8;          // C/D row offset for upper half-wave

  v8f acc[8] = {};                    // 8 N-tiles of 16 cols each

#ifdef HAS_WMMA_F32X4
  const int arow = min(row0 + m, nrows - 1);
  const float* __restrict__ xrow = X + (size_t)arow * NHID;
  const float* __restrict__ bcol = Ws + (m << 1);   // + kp*256 + t*32
#pragma unroll 4
  for (int k0 = 0; k0 < NHID; k0 += 4) {
    const v2f a = *(const v2f*)(xrow + k0 + khalf);          // global b64
    const float* __restrict__ bp = bcol + (((k0 >> 1) + khv) << 8);
#pragma unroll
    for (int t = 0; t < 8; ++t) {
      const v2f b = *(const v2f*)(bp + t * 32);              // ds_load_b64
      acc[t] = __builtin_amdgcn_wmma_f32_16x16x4_f32(
          false, a, false, b, (short)0, acc[t], false, false);
    }
  }
#else
  // scalar fallback with identical output mapping (packed LDS layout)
#pragma unroll
  for (int t = 0; t < 8; ++t)
    for (int v = 0; v < 8; ++v) {
      const int r = min(row0 + v + mhi, nrows - 1);
      const float* __restrict__ xr = X + (size_t)r * NHID;
      float s = 0.f;
      for (int k = 0; k < NHID; ++k)
        s += xr[k] * Ws[((k >> 1) << 8) + (((t << 4) + m) << 1) + (k & 1)];
      acc[t][v] = s;
    }
#endif

#pragma unroll
  for (int t = 0; t < 8; ++t)
#pragma unroll
    for (int v = 0; v < 8; ++v) {
      const int r = row0 + v + mhi;
      if (r < nrows) H[(size_t)r * NHID + t * 16 + m] = acc[t][v];
    }
}

// ---------------------------------------------------------------------------
// Attention logits: as[i] = h[i] . a_src ; ad[i] = h[i] . a_dst
// One wave per row, lane handles a float4 slice, shfl_xor tree reduce.
// ---------------------------------------------------------------------------
__global__ __launch_bounds__(256) void gat_alpha(
    const float* __restrict__ H, const float* __restrict__ a_src,
    const float* __restrict__ a_dst, float* __restrict__ as,
    float* __restrict__ ad, int nrows) {
  const int wid  = (int)((blockIdx.x * (size_t)blockDim.x + threadIdx.x) >> 5);
  const int lane = threadIdx.x & 31;
  if (wid >= nrows) return;
  const float4 hv = *(const float4*)(H + (size_t)wid * NHID + lane * 4);
  const float4 s4 = *(const float4*)(a_src + lane * 4);
  const float4 d4 = *(const float4*)(a_dst + lane * 4);
  float s = hv.x * s4.x + hv.y * s4.y + hv.z * s4.z + hv.w * s4.w;
  float d = hv.x * d4.x + hv.y * d4.y + hv.z * d4.z + hv.w * d4.w;
#pragma unroll
  for (int off = 16; off; off >>= 1) {
    s += __shfl_xor(s, off, 32);
    d += __shfl_xor(d, off, 32);
  }
  if (lane == 0) { as[wid] = s; ad[wid] = d; }
}

// ---------------------------------------------------------------------------
// Init accumulators: acc = 0, m = -inf, denom = 0 (d_out/ws are poisoned).
// ---------------------------------------------------------------------------
__global__ void gat_init(float* __restrict__ acc, float* __restrict__ m,
                         float* __restrict__ denom, int n, size_t total) {
  const size_t i = blockIdx.x * (size_t)blockDim.x + threadIdx.x;
  if (i < total) acc[i] = 0.f;
  if (i < (size_t)n) { m[i] = -3.402823466e38f; denom[i] = 0.f; }
}

__device__ __forceinline__ void atomicMaxF(float* addr, float v) {
  if (v >= 0.f) atomicMax((int*)addr, __float_as_int(v));
  else          atomicMin((unsigned int*)addr, __float_as_uint(v));
}

__device__ __forceinline__ float lrelu(float v, float slope) {
  return v >= 0.f ? v : slope * v;
}

// Pass 1: segment max of attention logits per destination (incl. self loops).
__global__ void gat_edge_max(const int* __restrict__ srcs,
                             const int* __restrict__ dsts,
                             const float* __restrict__ as,
                             const float* __restrict__ ad,
                             float* __restrict__ m, int nedges, int n) {
  const int e = (int)(blockIdx.x * (size_t)blockDim.x + threadIdx.x);
  if (e >= nedges + n) return;
  const int s = (e < nedges) ? srcs[e] : (e - nedges);
  const int d = (e < nedges) ? dsts[e] : (e - nedges);
  atomicMaxF(&m[d], lrelu(as[s] + ad[d], NEG_ATT));
}

// Pass 2: denom[dst] += exp(e - m[dst])
__global__ void gat_edge_den(const int* __restrict__ srcs,
                             const int* __restrict__ dsts,
                             const float* __restrict__ as,
                             const float* __restrict__ ad,
                             const float* __restrict__ m,
                             float* __restrict__ denom, int nedges, int n) {
  const int e = (int)(blockIdx.x * (size_t)blockDim.x + threadIdx.x);
  if (e >= nedges + n) return;
  const int s = (e < nedges) ? srcs[e] : (e - nedges);
  const int d = (e < nedges) ? dsts[e] : (e - nedges);
  const float v = lrelu(as[s] + ad[d], NEG_ATT);
  atomicAdd(&denom[d], expf(v - m[d]));
}

// Pass 3 (dominant): acc[dst] += h[src] * alpha.  One wave32 per edge,
// each lane carries a float4 slice: coalesced 512B gather + 4x f32 atomics.
__global__ __launch_bounds__(256) void gat_edge_aggr(
    const int* __restrict__ srcs, const int* __restrict__ dsts,
    const float* __restrict__ as, const float* __restrict__ ad,
    const float* __restrict__ m, const float* __restrict__ denom,
    const float* __restrict__ H, float* __restrict__ acc, int nedges, int n) {
  const long long wid = (long long)((blockIdx.x * (size_t)blockDim.x + threadIdx.x) >> 5);
  const int lane = threadIdx.x & 31;
  if (wid >= (long long)nedges + n) return;
  const int s = (wid < nedges) ? srcs[wid] : (int)(wid - nedges);
  const int d = (wid < nedges) ? dsts[wid] : (int)(wid - nedges);
  const float v = lrelu(as[s] + ad[d], NEG_ATT);
  const float w = expf(v - m[d]) / (denom[d] + 1e-16f);
  const float4 hv = *(const float4*)(H + (size_t)s * NHID + lane * 4);
  float* o = acc + (size_t)d * NHID + lane * 4;
  atomicAdd(o + 0, hv.x * w);
  atomicAdd(o + 1, hv.y * w);
  atomicAdd(o + 2, hv.z * w);
  atomicAdd(o + 3, hv.w * w);
}

// Finalize: io += bias, optional leaky_relu (between layers).
__global__ void gat_bias_act(float* __restrict__ io, const float* __restrict__ b,
                             size_t total, float slope, int apply_act) {
  const size_t i = blockIdx.x * (size_t)blockDim.x + threadIdx.x;
  if (i >= total) return;
  float v = io[i] + b[i & (NHID - 1)];
  if (apply_act) v = lrelu(v, slope);
  io[i] = v;
}

// ---------------------------------------------------------------------------
extern "C" void kernel_launch(void* const* d_in, const int* in_sizes, int n_in,
                              void* d_out, int out_size, void* d_ws, size_t ws_size,
                              hipStream_t stream) {
  const float* x    = (const float*)d_in[0];
  const int*   ei   = (const int*)d_in[1];
  const float* W1   = (const float*)d_in[2];
  const float* asr1 = (const float*)d_in[3];
  const float* adt1 = (const float*)d_in[4];
  const float* b1   = (const float*)d_in[5];
  const float* W2   = (const float*)d_in[6];
  const float* asr2 = (const float*)d_in[7];
  const float* adt2 = (const float*)d_in[8];
  const float* b2   = (const float*)d_in[9];
  float* out = (float*)d_out;

  const int N = in_sizes[0] / NHID;   // 50000
  const int E = in_sizes[1] / 2;      // 1600000
  const size_t NF = (size_t)N * NHID;

  float* h   = (float*)d_ws;          // [N,128]
  float* acc = h + NF;                // [N,128] layer-1 accum, then x2
  float* as  = acc + NF;              // [N]
  float* ad  = as + N;                // [N]
  float* mx  = ad + N;                // [N]
  float* den = mx + N;                // [N]

  const int* src = ei;
  const int* dst = ei + E;

  const int gGemm  = (N + 127) / 128;
  const int gWaveN = (int)(((size_t)N * 32 + 255) / 256);
  const int gNF    = (int)((NF + 255) / 256);
  const int gEdge  = (E + N + 255) / 256;
  const int gEdgeW = (int)(((size_t)(E + N) * 32 + 255) / 256);

  // ---- layer 1 ----
  gat_gemm128<<<gGemm, 256, 0, stream>>>(x, W1, h, N);
  gat_alpha<<<gWaveN, 256, 0, stream>>>(h, asr1, adt1, as, ad, N);
  gat_init<<<gNF, 256, 0, stream>>>(acc, mx, den, N, NF);
  gat_edge_max<<<gEdge, 256, 0, stream>>>(src, dst, as, ad, mx, E, N);
  gat_edge_den<<<gEdge, 256, 0, stream>>>(src, dst, as, ad, mx, den, E, N);
  gat_edge_aggr<<<gEdgeW, 256, 0, stream>>>(src, dst, as, ad, mx, den, h, acc, E, N);
  gat_bias_act<<<gNF, 256, 0, stream>>>(acc, b1, NF, NEG_ACT, 1);  // acc -> x2

  // ---- layer 2 ----
  gat_gemm128<<<gGemm, 256, 0, stream>>>(acc, W2, h, N);
  gat_alpha<<<gWaveN, 256, 0, stream>>>(h, asr2, adt2, as, ad, N);
  gat_init<<<gNF, 256, 0, stream>>>(out, mx, den, N, NF);
  gat_edge_max<<<gEdge, 256, 0, stream>>>(src, dst, as, ad, mx, E, N);
  gat_edge_den<<<gEdge, 256, 0, stream>>>(src, dst, as, ad, mx, den, E, N);
  gat_edge_aggr<<<gEdgeW, 256, 0, stream>>>(src, dst, as, ad, mx, den, h, out, E, N);
  gat_bias_act<<<gNF, 256, 0, stream>>>(out, b2, NF, 0.f, 0);
}